// BiAttentionLayer_16303695856109
// MI455X (gfx1250) — compile-verified
//
#include <hip/hip_runtime.h>
#include <hip/hip_bf16.h>

typedef __attribute__((ext_vector_type(2))) float v2f;
typedef __attribute__((ext_vector_type(8))) float v8f;

#define D2     200
#define TT     1024
#define JJ     64
#define NB     64
#define NEGV   -10000000.0f
#define UPITCH 208          // U padded to 13 d-tiles of 16 (zeros in 200..207)
#define TROWS  128          // t-rows per workgroup (8 waves x 16-row tile)

// ---------------------------------------------------------------------------
// Kernel 1: per (n, t-chunk): S GEMM (f32 WMMA), softmax over j, U_ GEMM,
// writes G blocks [200:400]=U_ and [400:600]=H*U_, stashes S_max to workspace.
// ---------------------------------------------------------------------------
__global__ __launch_bounds__(256)
void bidaf_k1_score_attend(const float* __restrict__ H,
                           const float* __restrict__ U,
                           const float* __restrict__ qmask,
                           const float* __restrict__ w,
                           const float* __restrict__ bptr,
                           float* __restrict__ out,
                           float* __restrict__ smax_ws)
{
    extern __shared__ float smem[];
    float* sU  = smem;               // 64*208 = 13312
    float* sW  = sU + 64 * UPITCH;   // 600 (w_h | w_u | w_hu)
    float* sQm = sW + 600;           // 64
    float* sHw = sQm + 64;           // 128  (H @ w_h for this chunk's rows)
    float* sUw = sHw + TROWS;        // 64   (U @ w_u)
    float* sSt = sUw + 64;           // 8 waves * 16*64 S/S_t tiles = 8192

    const int n     = blockIdx.x;
    const int tc    = blockIdx.y;
    const int tid   = threadIdx.x;
    const int tbase = tc * TROWS;
    const float bval = bptr[0];
    const float* Hn = H + (size_t)n * TT * D2;

    // ---- stage w, q_mask, U (zero-padded pitch) into LDS ----
    for (int i = tid; i < 600; i += 256) sW[i] = w[i];
    for (int i = tid; i < JJ; i += 256)  sQm[i] = qmask[n * JJ + i];
    for (int i = tid; i < JJ * D2; i += 256) {
        int j = i / D2, d = i - j * D2;
        sU[j * UPITCH + d] = U[((size_t)n * JJ + j) * D2 + d];
    }
    for (int i = tid; i < JJ * (UPITCH - D2); i += 256) {
        int j = i / (UPITCH - D2), d = D2 + (i - j * (UPITCH - D2));
        sU[j * UPITCH + d] = 0.0f;
    }
    __syncthreads();

    // ---- Hw for this chunk's rows, Uw for all j ----
    for (int r = tid; r < TROWS; r += 256) {
        const float* hr = Hn + (size_t)(tbase + r) * D2;
        float acc = 0.0f;
        for (int d = 0; d < D2; ++d) acc += hr[d] * sW[d];
        sHw[r] = acc;
    }
    for (int j = tid; j < JJ; j += 256) {
        float acc = 0.0f;
        for (int d = 0; d < D2; ++d) acc += sU[j * UPITCH + d] * sW[D2 + d];
        sUw[j] = acc;
    }
    __syncthreads();

    const int wave = tid >> 5;
    const int lane = tid & 31;
    const int lm   = lane & 15;   // row/col within 16
    const int lh   = lane >> 4;   // half-wave selector (K pair / M+8)
    float* St = sSt + wave * (16 * 64);
    const int t0 = tbase + wave * 16;          // one 16-row tile per wave

    // ================= S = (H .* w_hu) @ U^T  (16x64 tile) =================
    v8f c[4] = {v8f{}, v8f{}, v8f{}, v8f{}};
    const float* Arow = Hn + (size_t)(t0 + lm) * D2 + 2 * lh;
    const float* whu  = sW + 2 * D2;
    for (int k0 = 0; k0 < D2; k0 += 4) {
        v2f a = *(const v2f*)(Arow + k0);
        a.x *= whu[k0 + 2 * lh];
        a.y *= whu[k0 + 2 * lh + 1];
        const float* Bb = sU + lm * UPITCH + k0 + 2 * lh;
        v2f b0 = *(const v2f*)(Bb);
        v2f b1 = *(const v2f*)(Bb + 16 * UPITCH);
        v2f b2 = *(const v2f*)(Bb + 32 * UPITCH);
        v2f b3 = *(const v2f*)(Bb + 48 * UPITCH);
        c[0] = __builtin_amdgcn_wmma_f32_16x16x4_f32(false, a, false, b0, (short)0, c[0], false, false);
        c[1] = __builtin_amdgcn_wmma_f32_16x16x4_f32(false, a, false, b1, (short)0, c[1], false, false);
        c[2] = __builtin_amdgcn_wmma_f32_16x16x4_f32(false, a, false, b2, (short)0, c[2], false, false);
        c[3] = __builtin_amdgcn_wmma_f32_16x16x4_f32(false, a, false, b3, (short)0, c[3], false, false);
    }

    // D layout: lane, vgpr r -> row = r + 8*lh, col = lm (per 16-wide j tile)
#pragma unroll
    for (int jt = 0; jt < 4; ++jt)
#pragma unroll
        for (int r = 0; r < 8; ++r) {
            int row = r + 8 * lh;
            St[row * 64 + jt * 16 + lm] =
                c[jt][r] + sHw[wave * 16 + row] + sUw[jt * 16 + lm] + bval;
        }

    // ---- masked softmax over j (row = lm; both halves duplicate work) ----
    {
        float* Sr = St + lm * 64;
        float mx = -3.4e38f, smx = -3.4e38f;
        for (int j = 0; j < JJ; ++j) {
            float s = Sr[j], m = sQm[j];
            mx  = fmaxf(mx, s * m);
            smx = fmaxf(smx, s * m + NEGV * (1.0f - m));
        }
        float sum = 0.0f;
        for (int j = 0; j < JJ; ++j) {
            float s = Sr[j], m = sQm[j];
            float p = m * __expf(s * m - mx);
            Sr[j] = p;
            sum += p;
        }
        float inv = 1.0f / (sum + 1e-13f);
        for (int j = 0; j < JJ; ++j) Sr[j] *= inv;
        if (lh == 0) smax_ws[n * TT + t0 + lm] = smx;
    }

    // ================= U_ = S_t @ U  (16x200, 13 d-tiles) =================
    for (int dt = 0; dt < 13; ++dt) {
        const int d0 = dt * 16;
        v8f u = {};
        for (int k0 = 0; k0 < JJ; k0 += 4) {
            v2f a = *(const v2f*)(St + lm * 64 + k0 + 2 * lh);
            v2f b;
            b.x = sU[(k0 + 2 * lh) * UPITCH + d0 + lm];
            b.y = sU[(k0 + 2 * lh + 1) * UPITCH + d0 + lm];
            u = __builtin_amdgcn_wmma_f32_16x16x4_f32(false, a, false, b, (short)0, u, false, false);
        }
        const int d = d0 + lm;
        if (d < D2) {
#pragma unroll
            for (int r = 0; r < 8; ++r) {
                int t = t0 + r + 8 * lh;
                float uv = u[r];
                float hv = Hn[(size_t)t * D2 + d];
                size_t ob = ((size_t)n * TT + t) * (4 * D2);
                out[ob + D2 + d]     = uv;       // G block 1: U_
                out[ob + 2 * D2 + d] = hv * uv;  // G block 2: H * U_
            }
        }
    }
}

// ---------------------------------------------------------------------------
// Kernel 2: per n: a = masked_softmax(S_max, c_mask); H_ = sum_t a[t]*H[t,:]
// ---------------------------------------------------------------------------
__global__ __launch_bounds__(256)
void bidaf_k2_context(const float* __restrict__ H,
                      const float* __restrict__ cmask,
                      const float* __restrict__ smax_ws,
                      float* __restrict__ hbar_ws)
{
    __shared__ float sRed[256];
    __shared__ float sScal[2];
    __shared__ float sA[TT];
    const int n = blockIdx.x, tid = threadIdx.x;
    const float* Sm = smax_ws + n * TT;
    const float* Cm = cmask + (size_t)n * TT;

    float pm = -3.4e38f;
    for (int t = tid; t < TT; t += 256) pm = fmaxf(pm, Sm[t] * Cm[t]);
    sRed[tid] = pm;
    __syncthreads();
    if (tid == 0) {
        float M = -3.4e38f;
        for (int i = 0; i < 256; ++i) M = fmaxf(M, sRed[i]);
        sScal[0] = M;
    }
    __syncthreads();
    const float M = sScal[0];
    float ps = 0.0f;
    for (int t = tid; t < TT; t += 256) {
        float m = Cm[t];
        float p = m * __expf(Sm[t] * m - M);
        sA[t] = p;
        ps += p;
    }
    __syncthreads();
    sRed[tid] = ps;
    __syncthreads();
    if (tid == 0) {
        float s = 0.0f;
        for (int i = 0; i < 256; ++i) s += sRed[i];
        sScal[1] = 1.0f / (s + 1e-13f);
    }
    __syncthreads();
    const float inv = sScal[1];

    const float* Hn = H + (size_t)n * TT * D2;
    for (int d = tid; d < D2; d += 256) {
        float acc = 0.0f;
        for (int t = 0; t < TT; ++t) acc += sA[t] * Hn[(size_t)t * D2 + d];
        hbar_ws[n * D2 + d] = acc * inv;
    }
}

// ---------------------------------------------------------------------------
// Kernel 3: write G blocks 0 (H) and 3 (H * H_)
// ---------------------------------------------------------------------------
__global__ __launch_bounds__(256)
void bidaf_k3_write(const float* __restrict__ H,
                    const float* __restrict__ hbar_ws,
                    float* __restrict__ out)
{
    __shared__ float sHb[D2];
    const int n = blockIdx.x, tc = blockIdx.y, tid = threadIdx.x;
    for (int d = tid; d < D2; d += 256) sHb[d] = hbar_ws[n * D2 + d];
    __syncthreads();
    const float* Hn = H + (size_t)n * TT * D2 + (size_t)tc * TROWS * D2;
    for (int idx = tid; idx < TROWS * D2; idx += 256) {
        int t = idx / D2, d = idx - t * D2;
        float h = Hn[idx];
        size_t ob = ((size_t)n * TT + tc * TROWS + t) * (4 * D2);
        out[ob + d]          = h;            // G block 0: H
        out[ob + 3 * D2 + d] = h * sHb[d];   // G block 3: H * H_
    }
}

// ---------------------------------------------------------------------------
extern "C" void kernel_launch(void* const* d_in, const int* in_sizes, int n_in,
                              void* d_out, int out_size, void* d_ws, size_t ws_size,
                              hipStream_t stream)
{
    const float* H     = (const float*)d_in[0];
    const float* U     = (const float*)d_in[1];
    const float* cmask = (const float*)d_in[2];
    const float* qmask = (const float*)d_in[3];
    const float* w     = (const float*)d_in[4];
    const float* b     = (const float*)d_in[5];
    float* out = (float*)d_out;

    float* smax = (float*)d_ws;        // NB*TT floats
    float* hbar = smax + NB * TT;      // NB*D2 floats

    const size_t smem1 =
        (size_t)(64 * UPITCH + 600 + 64 + TROWS + 64 + 8 * 16 * 64) * sizeof(float);

    dim3 grid1(NB, TT / TROWS);        // 64 x 8 = 512 workgroups
    bidaf_k1_score_attend<<<grid1, 256, smem1, stream>>>(H, U, qmask, w, b, out, smax);
    bidaf_k2_context<<<NB, 256, 0, stream>>>(H, cmask, smax, hbar);
    bidaf_k3_write<<<grid1, 256, 0, stream>>>(H, hbar, out);
}